// QNN_0_Sum_7602092113980
// MI455X (gfx1250) — compile-verified
//
#include <hip/hip_runtime.h>
#include <math.h>

// ---------------------------------------------------------------------------
// QNN fused kernel for MI455X (gfx1250, wave32).
//
// Per item: closed-form exp(i*H) for the traceless Hermitian H built from the
// fixed SU(3) generators (lambda2, lambda5, lambda7, lambda3), then an 8-layer
// chain  v <- U_var[l] * (U_enc * v).  The shared U_var application is done as
// a real 6x6 GEMM over 32 items per wave using V_WMMA_F32_16X16X4_F32
// (two K-chunks of 4 accumulated through C; two 16-column groups).
// ---------------------------------------------------------------------------

typedef __attribute__((ext_vector_type(2))) float v2f;
typedef __attribute__((ext_vector_type(8))) float v8f;

struct c32 { float x, y; };

__device__ __forceinline__ c32 cmul(c32 a, c32 b) {
    c32 r; r.x = a.x*b.x - a.y*b.y; r.y = a.x*b.y + a.y*b.x; return r;
}
__device__ __forceinline__ c32 cmac(c32 acc, c32 a, c32 b) {
    acc.x = fmaf(a.x, b.x, fmaf(-a.y, b.y, acc.x));
    acc.y = fmaf(a.x, b.y, fmaf( a.y, b.x, acc.y));
    return acc;
}

// exp(i*H) for H = [[b3, -i b0, -i b1],[i b0, -b3, -i b2],[i b1, i b2, 0]].
// Traceless Hermitian: char poly lambda^3 = c1*lambda + q with
//   c1 = b0^2+b1^2+b2^2+b3^2,  q = det(H) = b3*(b1^2 - b2^2).
// exp(iH) = sum_k e^{i lam_k} (H^2 + lam_k H + (lam_k^2 - c1) I) / (3 lam_k^2 - c1)
//         = alpha*I + beta*H + gamma*H^2   (alpha,beta,gamma complex scalars).
__device__ void expiH(float b0, float b1, float b2, float b3, c32 U[9]) {
    float c1 = fmaf(b0, b0, fmaf(b1, b1, fmaf(b2, b2, b3*b3)));
    if (c1 < 1e-14f) {
        // H ~ 0 -> U = I
        #pragma unroll
        for (int i = 0; i < 9; ++i) { U[i].x = 0.f; U[i].y = 0.f; }
        U[0].x = 1.f; U[4].x = 1.f; U[8].x = 1.f;
        return;
    }
    float q = b3 * (b1*b1 - b2*b2);
    float s = sqrtf(c1 * (1.0f/3.0f));          // sqrt(c1/3)
    float ca = q / (2.0f * s * s * s);          // cos(3t)
    ca = fminf(1.0f, fmaxf(-1.0f, ca));
    float phi = acosf(ca) * (1.0f/3.0f);
    const float T23 = 2.0943951023931953f;      // 2*pi/3
    float lam0 = 2.f*s*cosf(phi);
    float lam1 = 2.f*s*cosf(phi - T23);
    float lam2 = 2.f*s*cosf(phi - 2.f*T23);

    c32 H[9] = {
        { b3, 0.f}, {0.f, -b0}, {0.f, -b1},
        {0.f,  b0}, {-b3, 0.f}, {0.f, -b2},
        {0.f,  b1}, {0.f,  b2}, {0.f, 0.f}
    };
    c32 H2[9];
    #pragma unroll
    for (int i = 0; i < 3; ++i)
        #pragma unroll
        for (int j = 0; j < 3; ++j) {
            c32 acc = cmul(H[i*3+0], H[0*3+j]);
            acc = cmac(acc, H[i*3+1], H[1*3+j]);
            acc = cmac(acc, H[i*3+2], H[2*3+j]);
            H2[i*3+j] = acc;
        }

    c32 A = {0.f,0.f}, Bc = {0.f,0.f}, G = {0.f,0.f};
    float lams[3] = {lam0, lam1, lam2};
    #pragma unroll
    for (int k = 0; k < 3; ++k) {
        float lk = lams[k];
        float d = fmaf(3.f*lk, lk, -c1);        // p'(lam) = 3 lam^2 - c1
        float ad = fabsf(d);
        d = (ad < 1e-9f) ? ((d < 0.f) ? -1e-9f : 1e-9f) : d;
        float inv = 1.0f / d;
        float sl = sinf(lk), cl = cosf(lk);
        c32 w = {cl*inv, sl*inv};
        G.x += w.x;               G.y += w.y;
        Bc.x = fmaf(w.x, lk, Bc.x); Bc.y = fmaf(w.y, lk, Bc.y);
        float t = fmaf(lk, lk, -c1);
        A.x = fmaf(w.x, t, A.x);  A.y = fmaf(w.y, t, A.y);
    }
    #pragma unroll
    for (int i = 0; i < 9; ++i) {
        c32 u = cmul(G, H2[i]);
        u = cmac(u, Bc, H[i]);
        U[i] = u;
    }
    U[0].x += A.x; U[0].y += A.y;
    U[4].x += A.x; U[4].y += A.y;
    U[8].x += A.x; U[8].y += A.y;
}

// Setup: 8 variational unitaries -> real 6x6 embeddings W6[8][6][6] in d_ws.
__global__ void qnn_setup(const float* __restrict__ weights, float* __restrict__ W6) {
    int l = threadIdx.x;
    if (l >= 8) return;
    c32 U[9];
    expiH(weights[l*4+0], weights[l*4+1], weights[l*4+2], weights[l*4+3], U);
    float* w = W6 + l*36;
    #pragma unroll
    for (int i = 0; i < 3; ++i)
        #pragma unroll
        for (int j = 0; j < 3; ++j) {
            float re = U[i*3+j].x, im = U[i*3+j].y;
            w[ i   *6 +  j     ] =  re;   // [[Re, -Im],
            w[ i   *6 + (j+3)  ] = -im;   //  [Im,  Re]]
            w[(i+3)*6 +  j     ] =  im;
            w[(i+3)*6 + (j+3)  ] =  re;
        }
}

__global__ __launch_bounds__(256)
void qnn_fused(const float* __restrict__ batch, const float* __restrict__ W6,
               float* __restrict__ out, int n) {
    const int lane = threadIdx.x & 31;
    const int wave = (blockIdx.x * blockDim.x + threadIdx.x) >> 5;
    const int item = wave * 32 + lane;
    const int li   = (item < n) ? item : (n - 1);   // keep full wave active
    const int lo   = lane & 15;
    const bool hi  = lane >= 16;

    const float4 bv = *(const float4*)(batch + (size_t)li * 4);
    c32 U[9];
    expiH(bv.x, bv.y, bv.z, bv.w, U);

    // state = U_enc * e0  (first encoding application); real embedding
    float s0 = U[0].x, s1 = U[3].x, s2 = U[6].x;   // Re v
    float s3 = U[0].y, s4 = U[3].y, s5 = U[6].y;   // Im v

    // Preload WMMA A-operands for all 8 layers.
    // A (16x4 f32): VGPR0 = K0(lanes0-15)/K2(lanes16-31); VGPR1 = K1/K3.
    const int r6 = (lo < 6) ? lo : 0;
    v2f a0[8], a1[8];
    #pragma unroll
    for (int l = 0; l < 8; ++l) {
        const float* w = W6 + l*36 + r6*6;
        float k0 = w[0], k1 = w[1], k2 = w[2], k3 = w[3], k4 = w[4], k5 = w[5];
        if (lo >= 6) { k0 = k1 = k2 = k3 = k4 = k5 = 0.f; }
        a0[l][0] = hi ? k2 : k0;
        a0[l][1] = hi ? k3 : k1;
        a1[l][0] = hi ? 0.f : k4;   // K=6,7 columns are zero padding
        a1[l][1] = hi ? 0.f : k5;
    }

    const v8f z = {0.f,0.f,0.f,0.f,0.f,0.f,0.f,0.f};

    #pragma unroll
    for (int l = 0; l < 8; ++l) {
        // ---- pack B operands (4x16 f32: VGPR0 = K0/K2 halves, VGPR1 = K1/K3)
        // group 1: columns = items 0..15 of this wave (state lives in lanes 0..15)
        float t2 = __shfl(s2, lo);
        float t3 = __shfl(s3, lo);
        v2f b1c0, b1c1;
        b1c0[0] = hi ? t2 : s0;
        b1c0[1] = hi ? t3 : s1;
        b1c1[0] = hi ? 0.f : s4;
        b1c1[1] = hi ? 0.f : s5;
        // group 2: columns = items 16..31 (state lives in lanes 16..31)
        float u0 = __shfl(s0, lo + 16);
        float u1 = __shfl(s1, lo + 16);
        float u2 = __shfl(s2, lo + 16);
        float u3 = __shfl(s3, lo + 16);
        float u4 = __shfl(s4, lo + 16);
        float u5 = __shfl(s5, lo + 16);
        v2f b2c0, b2c1;
        b2c0[0] = hi ? u2 : u0;
        b2c0[1] = hi ? u3 : u1;
        b2c1[0] = hi ? 0.f : u4;
        b2c1[1] = hi ? 0.f : u5;

        // ---- S' = W6[l] * S via f32 WMMA, K = 6 in two chunks of 4
        v8f d1 = __builtin_amdgcn_wmma_f32_16x16x4_f32(false, a0[l], false, b1c0,
                                                       (short)0, z,  false, false);
        d1     = __builtin_amdgcn_wmma_f32_16x16x4_f32(false, a1[l], false, b1c1,
                                                       (short)0, d1, false, false);
        v8f d2 = __builtin_amdgcn_wmma_f32_16x16x4_f32(false, a0[l], false, b2c0,
                                                       (short)0, z,  false, false);
        d2     = __builtin_amdgcn_wmma_f32_16x16x4_f32(false, a1[l], false, b2c1,
                                                       (short)0, d2, false, false);

        // ---- unpack D (row r of item n sits in VGPR r, lane n for n<16)
        float g0 = __shfl(d2[0], lo);
        float g1 = __shfl(d2[1], lo);
        float g2 = __shfl(d2[2], lo);
        float g3 = __shfl(d2[3], lo);
        float g4 = __shfl(d2[4], lo);
        float g5 = __shfl(d2[5], lo);
        s0 = hi ? g0 : d1[0];
        s1 = hi ? g1 : d1[1];
        s2 = hi ? g2 : d1[2];
        s3 = hi ? g3 : d1[3];
        s4 = hi ? g4 : d1[4];
        s5 = hi ? g5 : d1[5];

        // ---- per-item encoding unitary (complex 3x3 matvec, per lane)
        if (l < 7) {
            c32 v0 = {s0, s3}, v1 = {s1, s4}, v2c = {s2, s5};
            c32 r0 = cmul(U[0], v0); r0 = cmac(r0, U[1], v1); r0 = cmac(r0, U[2], v2c);
            c32 r1 = cmul(U[3], v0); r1 = cmac(r1, U[4], v1); r1 = cmac(r1, U[5], v2c);
            c32 r2 = cmul(U[6], v0); r2 = cmac(r2, U[7], v1); r2 = cmac(r2, U[8], v2c);
            s0 = r0.x; s1 = r1.x; s2 = r2.x;
            s3 = r0.y; s4 = r1.y; s5 = r2.y;
        }
    }

    // probabilities -> clamped sqrt fidelity -> normalize
    float p0 = fmaf(s0, s0, s3*s3);
    float p1 = fmaf(s1, s1, s4*s4);
    float p2 = fmaf(s2, s2, s5*s5);
    const float TWO_SQRT_EPS = 2e-5f;           // 2*sqrt(1e-10)
    float f0 = sqrtf(fmaxf(p0, 1e-10f)) + TWO_SQRT_EPS; f0 *= f0;
    float f1 = sqrtf(fmaxf(p1, 1e-10f)) + TWO_SQRT_EPS; f1 *= f1;
    float f2 = sqrtf(fmaxf(p2, 1e-10f)) + TWO_SQRT_EPS; f2 *= f2;
    float inv = 1.0f / (f0 + f1 + f2);
    if (item < n) {
        out[(size_t)item*3 + 0] = f0 * inv;
        out[(size_t)item*3 + 1] = f1 * inv;
        out[(size_t)item*3 + 2] = f2 * inv;
    }
}

extern "C" void kernel_launch(void* const* d_in, const int* in_sizes, int n_in,
                              void* d_out, int out_size, void* d_ws, size_t ws_size,
                              hipStream_t stream) {
    (void)n_in; (void)out_size; (void)ws_size;
    const float* batch   = (const float*)d_in[0];   // [B,4] f32
    const float* weights = (const float*)d_in[1];   // [8,4] f32
    float* out = (float*)d_out;                     // [B,3] f32
    float* W6  = (float*)d_ws;                      // 8*36 floats scratch

    const int n = in_sizes[0] / 4;                  // B items

    qnn_setup<<<1, 8, 0, stream>>>(weights, W6);
    const int blocks = (n + 255) / 256;             // 8 waves/block, 32 items/wave
    qnn_fused<<<blocks, 256, 0, stream>>>(batch, W6, out, n);
}